// SelectiveVariableWiseMessagePassingBlock_52510270161513
// MI455X (gfx1250) — compile-verified
//
#include <hip/hip_runtime.h>
#include <math.h>

typedef __attribute__((ext_vector_type(2))) float v2f;
typedef __attribute__((ext_vector_type(4))) float v4f;
typedef __attribute__((ext_vector_type(8))) float v8f;

constexpr int NV  = 36;    // TEMP_DIM
constexpr int ND  = 128;   // HIDDEN
constexpr int NT  = 256;   // SEQ_T
constexpr int NB  = 32;    // BATCH
constexpr int TC  = 64;    // T-chunk for v5 materialization
constexpr int NC  = 4608;  // V*d
constexpr int NO1 = 512;   // 4*d

#define WMMA_F32(a, b, c) \
  __builtin_amdgcn_wmma_f32_16x16x4_f32(false, (a), false, (b), (short)0, (c), false, false)

// ---------------------------------------------------------------------------
// Single 16x16 fp32 WMMA tile (for small GEMMs).
// A K-contiguous (element (m,k) at A[m*asM+k]) -> b64 loads.
// BVEC=true requires B k-contiguous (element (k,n) at B[k + n*bsN]).
// GUARD: clamp + cndmask zero-fill (branch-free, EXEC stays all-ones).
// ---------------------------------------------------------------------------
template<bool GUARD, bool BVEC>
__device__ __forceinline__ v8f wmma_f32_tile(
    const float* __restrict__ A, long asM, int m0, int Mmax,
    const float* __restrict__ B, long bsK, long bsN, int n0, int Nmax,
    int K)
{
  const int lane = threadIdx.x & 31;
  const int lm   = lane & 15;
  const int half = lane >> 4;
  int row = m0 + lm;
  int col = n0 + lm;
  bool rok = true, cok = true;
  if (GUARD) {
    rok = row < Mmax;
    cok = col < Nmax;
    row = rok ? row : (Mmax - 1);
    col = cok ? col : (Nmax - 1);
  }
  const float* __restrict__ Arow = A + (long)row * asM;
  v8f acc = {};
#pragma unroll 4
  for (int k = 0; k < K; k += 4) {
    const int ka = k + half * 2;
    v2f a = *(const v2f*)(Arow + ka);
    v2f b;
    if (BVEC) {
      b = *(const v2f*)(B + (long)ka + (long)col * bsN);
    } else {
      b.x = B[(long)ka * bsK + (long)col * bsN];
      b.y = B[(long)(ka + 1) * bsK + (long)col * bsN];
    }
    if (GUARD) {
      a.x = rok ? a.x : 0.0f;  a.y = rok ? a.y : 0.0f;
      b.x = cok ? b.x : 0.0f;  b.y = cok ? b.y : 0.0f;
    }
    acc = WMMA_F32(a, b, acc);
  }
  return acc;
}

// ---------------------------------------------------------------------------
// 32x32 register-blocked fp32 WMMA (4 accumulators): A fragment reused over
// 2 N-tiles, B over 2 M-tiles -> 1 load dword per WMMA (both operands BVEC).
// A element (m,k) at A[m*asM + k]; B element (k,n) at B[k + n*bsN].
// No guards: caller guarantees exact divisibility.
// ---------------------------------------------------------------------------
__device__ __forceinline__ void wmma_f32_tile32(
    const float* __restrict__ A, long asM, int m0,
    const float* __restrict__ B, long bsN, int n0, int K,
    v8f& c00, v8f& c01, v8f& c10, v8f& c11)
{
  const int lane = threadIdx.x & 31;
  const int lm   = lane & 15;
  const int half = lane >> 4;
  const float* __restrict__ A0 = A + (long)(m0 + lm) * asM;
  const float* __restrict__ A1 = A + (long)(m0 + 16 + lm) * asM;
  const float* __restrict__ B0 = B + (long)(n0 + lm) * bsN;
  const float* __restrict__ B1 = B + (long)(n0 + 16 + lm) * bsN;
#pragma unroll 2
  for (int k = 0; k < K; k += 4) {
    const int ka = k + half * 2;
    v2f a0 = *(const v2f*)(A0 + ka);
    v2f a1 = *(const v2f*)(A1 + ka);
    v2f b0 = *(const v2f*)(B0 + ka);
    v2f b1 = *(const v2f*)(B1 + ka);
    c00 = WMMA_F32(a0, b0, c00);
    c01 = WMMA_F32(a0, b1, c01);
    c10 = WMMA_F32(a1, b0, c10);
    c11 = WMMA_F32(a1, b1, c11);
  }
}

// Store an 8-consecutive-m run of a D tile (+ per-m bias), vectorized.
__device__ __forceinline__ void store8_bias(float* __restrict__ dst, const v8f& a,
                                            const float* __restrict__ bias)
{
  v4f lo = {a[0] + bias[0], a[1] + bias[1], a[2] + bias[2], a[3] + bias[3]};
  v4f hi = {a[4] + bias[4], a[5] + bias[5], a[6] + bias[6], a[7] + bias[7]};
  *(v4f*)dst = lo;
  *(v4f*)(dst + 4) = hi;
}

// ---------------------------------------------------------------------------
// pe[t,b,dd] : sinusoidal positional encoding
// ---------------------------------------------------------------------------
__global__ __launch_bounds__(256) void k_pe(const float* __restrict__ Xtimes,
                                            float* __restrict__ pe, long total)
{
  long i = (long)blockIdx.x * blockDim.x + threadIdx.x;
  if (i >= total) return;
  int dd  = (int)(i % ND);
  long tb = i / ND;                 // t*NB + b
  float x  = Xtimes[tb];
  int ii   = (dd < 64) ? dd : dd - 64;
  float ts = __powf(500.0f, (float)ii * (1.0f / 63.0f));
  float s  = x / ts;
  pe[i] = (dd < 64) ? __sinf(s) : __cosf(s);
}

// ---------------------------------------------------------------------------
// Per (b,v): float mask (transposed), sum_t(m), sum_t(m*x)
// ---------------------------------------------------------------------------
__global__ __launch_bounds__(256) void k_maskstats(
    const float* __restrict__ Xtemp, const int* __restrict__ mask,
    float* __restrict__ mf, float* __restrict__ sm, float* __restrict__ smx)
{
  int b = blockIdx.x / NV, v = blockIdx.x % NV;
  int t = threadIdx.x;
  float mm = (float)mask[((long)t * NB + b) * NV + v];
  float xv = Xtemp[((long)t * NB + b) * NV + v];
  mf[((long)b * NV + v) * NT + t] = mm;
  __shared__ float s1[256], s2[256];
  s1[t] = mm; s2[t] = mm * xv;
  __syncthreads();
  for (int off = 128; off > 0; off >>= 1) {
    if (t < off) { s1[t] += s1[t + off]; s2[t] += s2[t + off]; }
    __syncthreads();
  }
  if (t == 0) { sm[b * NV + v] = s1[0]; smx[b * NV + v] = s2[0]; }
}

// ---------------------------------------------------------------------------
// emb(32 x 4608) = Xstatic(32x16) @ emb_W(16x4608) + emb_b    [WMMA, exact]
// ---------------------------------------------------------------------------
__global__ __launch_bounds__(128) void k_emb(
    const float* __restrict__ Xs, const float* __restrict__ eW,
    const float* __restrict__ eb, float* __restrict__ emb)
{
  int wid = (blockIdx.x * 128 + threadIdx.x) >> 5;   // 0..575
  int mi = wid / 288, ni = wid % 288;
  v8f acc = wmma_f32_tile<false, false>(Xs, 16, mi * 16, 32,
                                        eW, NC, 1, ni * 16, NC, 16);
  int lane = threadIdx.x & 31;
  int n  = ni * 16 + (lane & 15);
  int mb = mi * 16 + (lane >> 4) * 8;
  float bias = eb[n];
#pragma unroll
  for (int r = 0; r < 8; ++r)
    emb[(long)(mb + r) * NC + n] = acc[r] + bias;
}

// ---------------------------------------------------------------------------
// smpe[b](36x128) = mf_b(36x256) @ pe_b(256x128)              [WMMA, M guard]
// ---------------------------------------------------------------------------
__global__ __launch_bounds__(128) void k_smpe(
    const float* __restrict__ mf, const float* __restrict__ pe,
    float* __restrict__ smpe)
{
  int wid = (blockIdx.x * 128 + threadIdx.x) >> 5;   // 0..767
  int b = wid / 24, rr = wid % 24, mi = rr / 8, ni = rr % 8;
  v8f acc = wmma_f32_tile<true, false>(mf + (long)b * NV * NT, NT, mi * 16, NV,
                                       pe + (long)b * ND, (long)NB * ND, 1,
                                       ni * 16, ND, NT);
  int lane = threadIdx.x & 31;
  int n  = ni * 16 + (lane & 15);
  int mb = mi * 16 + (lane >> 4) * 8;
#pragma unroll
  for (int r = 0; r < 8; ++r) {
    int m = mb + r;
    if (m < NV) smpe[((long)b * NV + m) * ND + n] = acc[r];
  }
}

// ---------------------------------------------------------------------------
// h4[b,v,dd] = (temp_w*smx + (temp_b+emb)*sm + smpe) / (sm+1)
// ---------------------------------------------------------------------------
__global__ __launch_bounds__(128) void k_h4(
    const float* __restrict__ temp_w, const float* __restrict__ temp_b,
    const float* __restrict__ embf, const float* __restrict__ smpe,
    const float* __restrict__ sm, const float* __restrict__ smx,
    float* __restrict__ h4)
{
  int b = blockIdx.x / NV, v = blockIdx.x % NV;
  int dd = threadIdx.x;
  float smv = sm[b * NV + v], smxv = smx[b * NV + v];
  long iw = (long)v * ND + dd;
  float h2 = temp_w[iw] * smxv + (temp_b[iw] + embf[(long)b * NC + iw]) * smv
           + smpe[((long)b * NV + v) * ND + dd];
  h4[((long)b * NV + v) * ND + dd] = h2 / (smv + 1.0f);
}

// ---------------------------------------------------------------------------
// q/k (1152x128) = h4 @ {Wq,Wk} * scale ; mean_v stored TRANSPOSED:
// mvT[b][dd][v] so the attention epilogue bias loads vectorize.  [WMMA, exact]
// ---------------------------------------------------------------------------
__global__ __launch_bounds__(128) void k_qkv(
    const float* __restrict__ h4, const float* __restrict__ Wq,
    const float* __restrict__ Wk, const float* __restrict__ Wv,
    float* __restrict__ qb, float* __restrict__ kb, float* __restrict__ mvT)
{
  int wid = (blockIdx.x * 128 + threadIdx.x) >> 5;   // 0..1727
  int mat = wid / 576, rem = wid % 576, mi = rem / 8, ni = rem % 8;
  const float* Wm = (mat == 0) ? Wq : (mat == 1) ? Wk : Wv;
  float sc        = (mat == 2) ? 1.0f : 0.17677669529663687f;  // 1/sqrt(32)
  v8f acc = wmma_f32_tile<false, false>(h4, ND, mi * 16, NB * NV,
                                        Wm, ND, 1, ni * 16, ND, ND);
  int lane = threadIdx.x & 31;
  int n  = ni * 16 + (lane & 15);
  int mb = mi * 16 + (lane >> 4) * 8;
  if (mat == 2) {
#pragma unroll
    for (int r = 0; r < 8; ++r) {
      int m = mb + r, bq = m / NV, v = m % NV;
      mvT[((long)bq * ND + n) * NV + v] = acc[r];
    }
  } else {
    float* outp = (mat == 0) ? qb : kb;
#pragma unroll
    for (int r = 0; r < 8; ++r)
      outp[(long)(mb + r) * ND + n] = acc[r] * sc;
  }
}

// ---------------------------------------------------------------------------
// attA[b,h](48x48 padded) = q_bh(36x32) @ k_bh^T(32x36)  [WMMA, guards, BVEC]
// ---------------------------------------------------------------------------
__global__ __launch_bounds__(128) void k_scores(
    const float* __restrict__ qb, const float* __restrict__ kb,
    float* __restrict__ attA)
{
  int wid = (blockIdx.x * 128 + threadIdx.x) >> 5;   // 0..1151
  int bh = wid / 9, rr = wid % 9, mi = rr / 3, ni = rr % 3;
  int b = bh / 4, hh = bh % 4;
  const float* Aq = qb + (long)b * NV * ND + hh * 32;   // (v,kk): v*ND + kk
  const float* Bk = kb + (long)b * NV * ND + hh * 32;   // (kk,u): u*ND + kk
  v8f acc = wmma_f32_tile<true, true>(Aq, ND, mi * 16, NV,
                                      Bk, 1, ND, ni * 16, NV, 32);
  int lane = threadIdx.x & 31;
  int n  = ni * 16 + (lane & 15);
  int mb = mi * 16 + (lane >> 4) * 8;
#pragma unroll
  for (int r = 0; r < 8; ++r)   // padded entries are exact zeros
    attA[((long)bh * 48 + (mb + r)) * 48 + n] = acc[r];
}

// ---------------------------------------------------------------------------
// Attention apply epilogue: acc rows are consecutive c = dd*NV + (mb..mb+7)
// in t-major v5 -> vectorized stores, vectorized mean_v bias.
// ---------------------------------------------------------------------------
__device__ __forceinline__ void attn_store(const v8f& acc, int mb, int dd,
    const float* __restrict__ mvT, float* __restrict__ v5t, int b, int tl)
{
  if (mb >= NV) return;                               // mb==40 half: nothing
  const float* bias = mvT + ((long)b * ND + dd) * NV + mb;
  float* dst = v5t + ((long)b * TC + tl) * NC + dd * NV + mb;
  v4f lo = {acc[0] + bias[0], acc[1] + bias[1], acc[2] + bias[2], acc[3] + bias[3]};
  *(v4f*)dst = lo;
  if (mb + 8 <= NV) {                                 // full 8-run
    v4f hi = {acc[4] + bias[4], acc[5] + bias[5], acc[6] + bias[6], acc[7] + bias[7]};
    *(v4f*)(dst + 4) = hi;
  }
}

// ---------------------------------------------------------------------------
// Per (b,t): masked softmax, then S(48x40) @ H(40x32) per head as a single
// 48x32 register-blocked WMMA macro-tile (6 accumulators, b64 LDS loads).
// ---------------------------------------------------------------------------
__global__ __launch_bounds__(128) void k_attn(
    const float* __restrict__ Xtemp, const int* __restrict__ mask,
    const float* __restrict__ temp_w, const float* __restrict__ temp_b,
    const float* __restrict__ pe, const float* __restrict__ embf,
    const float* __restrict__ attA, const float* __restrict__ mvT,
    float* __restrict__ v5t, int t0)
{
  __shared__ float sS[4][48][40];     // [head][v][u-padded]
  __shared__ float sHT[4][32][40];    // [head][kk][u-padded]  (H transposed)
  int b  = blockIdx.x / TC;
  int tl = blockIdx.x % TC;
  int t  = t0 + tl;
  int hh   = threadIdx.x >> 5;
  int lane = threadIdx.x & 31;

  // H values for this head, transposed: sHT[kk][u] = H[hh*32+kk, u, t]
  for (int idx = lane; idx < 32 * 40; idx += 32) {
    int kk = idx / 40, u = idx % 40;
    int dd = hh * 32 + kk;
    float val = 0.0f;
    if (u < NV) {
      float xv = Xtemp[((long)t * NB + b) * NV + u];
      val = xv * temp_w[(long)u * ND + dd] + temp_b[(long)u * ND + dd]
          + pe[((long)t * NB + b) * ND + dd]
          + embf[(long)b * NC + (long)u * ND + dd];
    }
    sHT[hh][kk][u] = val;
  }
  // masked softmax rows
  for (int v = lane; v < 48; v += 32) {
    if (v < NV) {
      float rowv[NV];
      float mx = -3.4e38f;
#pragma unroll
      for (int u = 0; u < NV; ++u) {
        float a = attA[(((long)b * 4 + hh) * 48 + v) * 48 + u];
        float x = a + ((mask[((long)t * NB + b) * NV + u] != 0) ? 0.0f : -1.0e6f);
        rowv[u] = x;
        mx = fmaxf(mx, x);
      }
      float s = 0.0f;
#pragma unroll
      for (int u = 0; u < NV; ++u) { float e = __expf(rowv[u] - mx); rowv[u] = e; s += e; }
      float inv = 1.0f / s;
#pragma unroll
      for (int u = 0; u < 40; ++u) sS[hh][v][u] = (u < NV) ? rowv[u] * inv : 0.0f;
    } else {
#pragma unroll
      for (int u = 0; u < 40; ++u) sS[hh][v][u] = 0.0f;
    }
  }
  __syncthreads();

  const int lm = lane & 15, half = lane >> 4;
  v8f c00 = {}, c01 = {}, c10 = {}, c11 = {}, c20 = {}, c21 = {};
#pragma unroll
  for (int k = 0; k < 40; k += 4) {
    int ka = k + half * 2;
    v2f a0 = *(const v2f*)&sS[hh][lm][ka];
    v2f a1 = *(const v2f*)&sS[hh][lm + 16][ka];
    v2f a2 = *(const v2f*)&sS[hh][lm + 32][ka];
    v2f b0 = *(const v2f*)&sHT[hh][lm][ka];
    v2f b1 = *(const v2f*)&sHT[hh][lm + 16][ka];
    c00 = WMMA_F32(a0, b0, c00);  c01 = WMMA_F32(a0, b1, c01);
    c10 = WMMA_F32(a1, b0, c10);  c11 = WMMA_F32(a1, b1, c11);
    c20 = WMMA_F32(a2, b0, c20);  c21 = WMMA_F32(a2, b1, c21);
  }
  int n0 = lm, hb = half * 8;
  attn_store(c00,  0 + hb, hh * 32 + n0,      mvT, v5t, b, tl);
  attn_store(c01,  0 + hb, hh * 32 + n0 + 16, mvT, v5t, b, tl);
  attn_store(c10, 16 + hb, hh * 32 + n0,      mvT, v5t, b, tl);
  attn_store(c11, 16 + hb, hh * 32 + n0 + 16, mvT, v5t, b, tl);
  attn_store(c20, 32 + hb, hh * 32 + n0,      mvT, v5t, b, tl);
  attn_store(c21, 32 + hb, hh * 32 + n0 + 16, mvT, v5t, b, tl);
}

// ---------------------------------------------------------------------------
// y1 chunk: W1(512x4608) @ v5t_b^T -> t-major y1 [b][t][o]
// 32x32 register-blocked, both operands k-contiguous.        [WMMA, exact]
// ---------------------------------------------------------------------------
__global__ __launch_bounds__(128) void k_y1(
    const float* __restrict__ W1, const float* __restrict__ b1,
    const float* __restrict__ v5t, float* __restrict__ y1t, int t0)
{
  int wid = (blockIdx.x * 128 + threadIdx.x) >> 5;   // 0..1023
  int b = wid / 32, rr = wid % 32, mblk = rr / 2, nblk = rr % 2;
  int m0 = mblk * 32, n0 = nblk * 32;
  v8f c00 = {}, c01 = {}, c10 = {}, c11 = {};
  wmma_f32_tile32(W1, NC, m0, v5t + (long)b * TC * NC, NC, n0, NC,
                  c00, c01, c10, c11);
  const int lane = threadIdx.x & 31;
  const int lm = lane & 15, hb = (lane >> 4) * 8;
  long rowbase = ((long)b * NT + t0) * NO1;
  // tile (mt,nt): n = n0 + nt*16 + lm ; m-run starts at m0 + mt*16 + hb
  store8_bias(y1t + rowbase + (long)(n0 + lm) * NO1      + m0 + hb,      c00, b1 + m0 + hb);
  store8_bias(y1t + rowbase + (long)(n0 + 16 + lm) * NO1 + m0 + hb,      c01, b1 + m0 + hb);
  store8_bias(y1t + rowbase + (long)(n0 + lm) * NO1      + m0 + 16 + hb, c10, b1 + m0 + 16 + hb);
  store8_bias(y1t + rowbase + (long)(n0 + 16 + lm) * NO1 + m0 + 16 + hb, c11, b1 + m0 + 16 + hb);
}

// ---------------------------------------------------------------------------
// y2: W2(128x512) @ y1t_b^T -> t-major y2 [b][t][dd]         [WMMA, exact]
// ---------------------------------------------------------------------------
__global__ __launch_bounds__(128) void k_y2(
    const float* __restrict__ W2, const float* __restrict__ b2,
    const float* __restrict__ y1t, float* __restrict__ y2t)
{
  int wid = (blockIdx.x * 128 + threadIdx.x) >> 5;   // 0..1023
  int b = wid / 32, rr = wid % 32, mblk = rr / 8, nblk = rr % 8;
  int m0 = mblk * 32, n0 = nblk * 32;
  v8f c00 = {}, c01 = {}, c10 = {}, c11 = {};
  wmma_f32_tile32(W2, NO1, m0, y1t + (long)b * NT * NO1, NO1, n0, NO1,
                  c00, c01, c10, c11);
  const int lane = threadIdx.x & 31;
  const int lm = lane & 15, hb = (lane >> 4) * 8;
  long rowbase = (long)b * NT * ND;
  store8_bias(y2t + rowbase + (long)(n0 + lm) * ND      + m0 + hb,      c00, b2 + m0 + hb);
  store8_bias(y2t + rowbase + (long)(n0 + 16 + lm) * ND + m0 + hb,      c01, b2 + m0 + hb);
  store8_bias(y2t + rowbase + (long)(n0 + lm) * ND      + m0 + 16 + hb, c10, b2 + m0 + 16 + hb);
  store8_bias(y2t + rowbase + (long)(n0 + 16 + lm) * ND + m0 + 16 + hb, c11, b2 + m0 + 16 + hb);
}

// ---------------------------------------------------------------------------
// BatchNorm stats over (b,t) per channel; x is t-major [b][t][C]
// ---------------------------------------------------------------------------
__global__ __launch_bounds__(256) void k_bnstats(
    const float* __restrict__ x, float* __restrict__ stats, int C)
{
  int c = blockIdx.x;
  __shared__ float s1[256], s2[256];
  float s = 0.0f, q = 0.0f;
  for (int i = threadIdx.x; i < NB * NT; i += 256) {
    float v = x[(long)i * C + c];
    s += v; q += v * v;
  }
  s1[threadIdx.x] = s; s2[threadIdx.x] = q;
  __syncthreads();
  for (int off = 128; off > 0; off >>= 1) {
    if (threadIdx.x < off) { s1[threadIdx.x] += s1[threadIdx.x + off];
                             s2[threadIdx.x] += s2[threadIdx.x + off]; }
    __syncthreads();
  }
  if (threadIdx.x == 0) {
    float N = (float)(NB * NT);
    float mu = s1[0] / N;
    stats[c * 2] = mu;
    stats[c * 2 + 1] = s2[0] / N - mu * mu;
  }
}

// ---------------------------------------------------------------------------
// Apply BN + exact GELU in place (t-major: channel = i % C)
// ---------------------------------------------------------------------------
__global__ __launch_bounds__(256) void k_bngelu(
    float* __restrict__ x, const float* __restrict__ stats,
    const float* __restrict__ g, const float* __restrict__ bta,
    int C, long total)
{
  long i = (long)blockIdx.x * blockDim.x + threadIdx.x;
  if (i >= total) return;
  int c = (int)(i % C);
  float mu = stats[c * 2], var = stats[c * 2 + 1];
  float xn = (x[i] - mu) * rsqrtf(var + 1e-5f) * g[c] + bta[c];
  x[i] = 0.5f * xn * (1.0f + erff(xn * 0.70710678118654752f));
}

// ---------------------------------------------------------------------------
// Final LayerNorm over channels + residual; z t-major [b][t][dd],
// output row-major (B, d, T) as the reference returns.
// ---------------------------------------------------------------------------
__global__ __launch_bounds__(128) void k_ln(
    const float* __restrict__ z, const float* __restrict__ g,
    const float* __restrict__ bta, float* __restrict__ out)
{
  int b = blockIdx.x / NT, t = blockIdx.x % NT;
  int dd = threadIdx.x;
  float zv = z[(long)blockIdx.x * ND + dd];
  __shared__ float s1[128], s2[128];
  s1[dd] = zv; s2[dd] = zv * zv;
  __syncthreads();
  for (int off = 64; off > 0; off >>= 1) {
    if (dd < off) { s1[dd] += s1[dd + off]; s2[dd] += s2[dd + off]; }
    __syncthreads();
  }
  float mu = s1[0] * (1.0f / ND);
  float var = s2[0] * (1.0f / ND) - mu * mu;
  float zn = (zv - mu) * rsqrtf(var + 1e-5f) * g[dd] + bta[dd];
  out[((long)b * ND + dd) * NT + t] = zv + zn;
}

// ---------------------------------------------------------------------------
extern "C" void kernel_launch(void* const* d_in, const int* in_sizes, int n_in,
                              void* d_out, int out_size, void* d_ws, size_t ws_size,
                              hipStream_t stream)
{
  const float* Xtemp   = (const float*)d_in[0];
  const float* Xtimes  = (const float*)d_in[1];
  const float* Xstatic = (const float*)d_in[2];
  const int*   mask    = (const int*)d_in[3];
  const float* Wq      = (const float*)d_in[4];
  const float* Wk      = (const float*)d_in[5];
  const float* Wv      = (const float*)d_in[6];
  const float* temp_w  = (const float*)d_in[7];
  const float* temp_b  = (const float*)d_in[8];
  const float* emb_W   = (const float*)d_in[9];
  const float* emb_b   = (const float*)d_in[10];
  const float* W1      = (const float*)d_in[11];
  const float* b1      = (const float*)d_in[12];
  const float* bn1g    = (const float*)d_in[13];
  const float* bn1b    = (const float*)d_in[14];
  const float* W2      = (const float*)d_in[15];
  const float* b2      = (const float*)d_in[16];
  const float* bn2g    = (const float*)d_in[17];
  const float* bn2b    = (const float*)d_in[18];
  const float* lng     = (const float*)d_in[19];
  const float* lnb     = (const float*)d_in[20];
  float* out = (float*)d_out;
  float* ws  = (float*)d_ws;

  const long OFF_PE   = 0;
  const long OFF_MF   = OFF_PE   + (long)NT * NB * ND;
  const long OFF_SM   = OFF_MF   + (long)NB * NV * NT;
  const long OFF_SMX  = OFF_SM   + (long)NB * NV;
  const long OFF_EMB  = OFF_SMX  + (long)NB * NV;
  const long OFF_SMPE = OFF_EMB  + (long)NB * NC;
  const long OFF_H4   = OFF_SMPE + (long)NB * NV * ND;
  const long OFF_Q    = OFF_H4   + (long)NB * NV * ND;
  const long OFF_K    = OFF_Q    + (long)NB * NV * ND;
  const long OFF_MV   = OFF_K    + (long)NB * NV * ND;
  const long OFF_ATTA = OFF_MV   + (long)NB * NV * ND;
  const long OFF_V5C  = OFF_ATTA + (long)NB * 4 * 48 * 48;
  const long OFF_Y1   = OFF_V5C  + (long)NB * NC * TC;
  const long OFF_Y2   = OFF_Y1   + (long)NB * NO1 * NT;
  const long OFF_S1   = OFF_Y2   + (long)NB * ND * NT;
  const long OFF_S2   = OFF_S1   + 1024;
  const long OFF_END  = OFF_S2   + 256;
  if (ws_size < (size_t)OFF_END * sizeof(float)) return;

  float* pe   = ws + OFF_PE;
  float* mf   = ws + OFF_MF;
  float* sm   = ws + OFF_SM;
  float* smx  = ws + OFF_SMX;
  float* embf = ws + OFF_EMB;
  float* smpe = ws + OFF_SMPE;
  float* h4   = ws + OFF_H4;
  float* qb   = ws + OFF_Q;
  float* kb   = ws + OFF_K;
  float* mvT  = ws + OFF_MV;
  float* attA = ws + OFF_ATTA;
  float* v5t  = ws + OFF_V5C;
  float* y1t  = ws + OFF_Y1;
  float* y2t  = ws + OFF_Y2;
  float* st1  = ws + OFF_S1;
  float* st2  = ws + OFF_S2;

  k_pe<<<4096, 256, 0, stream>>>(Xtimes, pe, (long)NT * NB * ND);
  k_maskstats<<<NB * NV, 256, 0, stream>>>(Xtemp, mask, mf, sm, smx);
  k_emb<<<144, 128, 0, stream>>>(Xstatic, emb_W, emb_b, embf);
  k_smpe<<<192, 128, 0, stream>>>(mf, pe, smpe);
  k_h4<<<NB * NV, 128, 0, stream>>>(temp_w, temp_b, embf, smpe, sm, smx, h4);
  k_qkv<<<432, 128, 0, stream>>>(h4, Wq, Wk, Wv, qb, kb, mvT);
  k_scores<<<288, 128, 0, stream>>>(qb, kb, attA);
  for (int t0 = 0; t0 < NT; t0 += TC) {
    k_attn<<<NB * TC, 128, 0, stream>>>(Xtemp, mask, temp_w, temp_b, pe, embf,
                                        attA, mvT, v5t, t0);
    k_y1<<<256, 128, 0, stream>>>(W1, b1, v5t, y1t, t0);
  }
  k_bnstats<<<NO1, 256, 0, stream>>>(y1t, st1, NO1);
  k_bngelu<<<16384, 256, 0, stream>>>(y1t, st1, bn1g, bn1b, NO1, (long)NB * NO1 * NT);
  k_y2<<<256, 128, 0, stream>>>(W2, b2, y1t, y2t);
  k_bnstats<<<ND, 256, 0, stream>>>(y2t, st2, ND);
  k_bngelu<<<4096, 256, 0, stream>>>(y2t, st2, bn2g, bn2b, ND, (long)NB * ND * NT);
  k_ln<<<NB * NT, 128, 0, stream>>>(y2t, lng, lnb, out);
}